// WignerDRotation_7232724927062
// MI455X (gfx1250) — compile-verified
//
#include <hip/hip_runtime.h>

// ---------------------------------------------------------------------------
// Wigner-D rotation, fused:  out[n] = Z(a) J Z(b) J Z(g) x[n]
// Stage 1: per-batch composite block matrices D_l (l=1,2,3) -> d_ws
// Stage 2: single streaming pass, block-diag apply via v_wmma_f32_16x16x4_f32
// Memory-bound: 554 MB total traffic -> ~24 us floor at 23.3 TB/s.
// Workspace requirement: 1024 * 84 * 4 B = 344 KB.
// ---------------------------------------------------------------------------

typedef __attribute__((ext_vector_type(2))) float v2f;
typedef __attribute__((ext_vector_type(4))) float v4f;
typedef __attribute__((ext_vector_type(8))) float v8f;

#define NB   1024
#define DIM  2112
#define NCH  32

// J_l: real-SH matrix of the y<->z axis swap in this code's convention
// (Condon-Shortley phases included). Derived analytically; J = J^T = J^-1.
#define S6_4  0.6123724356957945f   // sqrt(6)/4
#define S10_4 0.7905694150420949f   // sqrt(10)/4
#define S15_4 0.9682458365518543f   // sqrt(15)/4
#define S3_2  0.8660254037844386f   // sqrt(3)/2

__constant__ float cJ1[9] = {
    0.f, -1.f, 0.f,
   -1.f,  0.f, 0.f,
    0.f,  0.f, 1.f };

__constant__ float cJ2[25] = {
    0.f, 0.f,  0.f , -1.f,  0.f ,
    0.f, 1.f,  0.f ,  0.f,  0.f ,
    0.f, 0.f, -0.5f,  0.f, -S3_2,
   -1.f, 0.f,  0.f ,  0.f,  0.f ,
    0.f, 0.f, -S3_2,  0.f,  0.5f };

__constant__ float cJ3[49] = {
    0.f  , 0.f, 0.f  ,  S10_4, 0.f   , -S6_4 , 0.f   ,
    0.f  , 1.f, 0.f  ,  0.f  , 0.f   ,  0.f  , 0.f   ,
    0.f  , 0.f, 0.f  ,  S6_4 , 0.f   ,  S10_4, 0.f   ,
    S10_4, 0.f, S6_4 ,  0.f  , 0.f   ,  0.f  , 0.f   ,
    0.f  , 0.f, 0.f  ,  0.f  , -0.25f,  0.f  , -S15_4,
   -S6_4 , 0.f, S10_4,  0.f  , 0.f   ,  0.f  , 0.f   ,
    0.f  , 0.f, 0.f  ,  0.f  , -S15_4,  0.f  ,  0.25f };

// ---------------------------------------------------------------------------
// Kernel A: per batch n compute D_l = Z(alpha) J Z(beta) J Z(gamma)
// Stored compactly: D1 @ +0 (9), D2 @ +9 (25), D3 @ +34 (49); stride 84.
// ---------------------------------------------------------------------------
__global__ __launch_bounds__(256) void wigner_build_D(
    const float* __restrict__ alpha, const float* __restrict__ beta,
    const float* __restrict__ gamma, float* __restrict__ Dws)
{
  int n = blockIdx.x * blockDim.x + threadIdx.x;
  if (n >= NB) return;
  float a = alpha[n], b = beta[n], g = gamma[n];
  float* out = Dws + (size_t)n * 84;

  float M[49], T[49];
  for (int li = 0; li < 3; ++li) {
    const int l = li + 1, d = 2 * l + 1;
    const float* J = (li == 0) ? cJ1 : (li == 1) ? cJ2 : cJ3;
    const int off = (li == 0) ? 0 : (li == 1) ? 9 : 34;

    // T = Z(gamma):  row p: +cos((p-l)g) at col p, -sin((p-l)g) at col d-1-p
    for (int i = 0; i < d * d; ++i) T[i] = 0.f;
    for (int p = 0; p < d; ++p) {
      float m = (float)(p - l);
      T[p * d + p]           += cosf(m * g);
      T[p * d + (d - 1 - p)] -= sinf(m * g);
    }
    // M = J * T
    for (int p = 0; p < d; ++p)
      for (int q = 0; q < d; ++q) {
        float s = 0.f;
        for (int k = 0; k < d; ++k) s += J[p * d + k] * T[k * d + q];
        M[p * d + q] = s;
      }
    // T = Z(beta) * M
    for (int p = 0; p < d; ++p) {
      float m = (float)(p - l);
      float cb = cosf(m * b), sb = sinf(m * b);
      for (int q = 0; q < d; ++q)
        T[p * d + q] = cb * M[p * d + q] - sb * M[(d - 1 - p) * d + q];
    }
    // M = J * T
    for (int p = 0; p < d; ++p)
      for (int q = 0; q < d; ++q) {
        float s = 0.f;
        for (int k = 0; k < d; ++k) s += J[p * d + k] * T[k * d + q];
        M[p * d + q] = s;
      }
    // T = Z(alpha) * M ; store
    for (int p = 0; p < d; ++p) {
      float m = (float)(p - l);
      float ca = cosf(m * a), sa = sinf(m * a);
      for (int q = 0; q < d; ++q)
        out[off + p * d + q] = ca * M[p * d + q] - sa * M[(d - 1 - p) * d + q];
    }
  }
}

// ---------------------------------------------------------------------------
// Kernel B: one workgroup per batch row. 16x16 block-diagonal tiles in LDS,
// streamed through v_wmma_f32_16x16x4_f32. l=0 is a b128 copy.
//   tile0 = blockdiag(D1 x4) rows 0..11, 64 chunks, 3 k-steps (K=12..15 dead)
//   tile1 = blockdiag(D2 x2) rows 0..9,  64 chunks, 3 k-steps (K=12..15 dead)
//   tile2 = blockdiag(D3 x2) rows 2..15, 32 chunks, 4 k-steps (shifted +2 so
//           padding UNDER-runs into the in-bounds l=2 section)
// Per wave: fixed channel half (wave&1), chunk start (wave>>1), chunk stride 4
// -> compile-time trip counts, all addresses immediate offsets.
// ---------------------------------------------------------------------------

template<int TILE, int MODE, int NTILES, int ROWSTEP, int NSTEPS>
__device__ __forceinline__ void run_section(
    const float (*sD)[16][16], const float* __restrict__ xb, float* __restrict__ yb,
    int secrow0, int wave, int laneN, int hi)
{
  // Hoisted A fragments: k-step s -> lanes 0-15 give K=4s+{0,1}, 16-31 K=4s+{2,3}
  v2f A[NSTEPS];
#pragma unroll
  for (int s = 0; s < NSTEPS; ++s) {
    A[s][0] = sD[TILE][laneN][4 * s + 2 * hi + 0];
    A[s][1] = sD[TILE][laneN][4 * s + 2 * hi + 1];
  }

  const int ch   = (wave & 1) * 16 + laneN;           // per-wave constant
  const int wrow = secrow0 + (wave >> 1) * ROWSTEP;   // per-wave chunk start
  const float* p = xb + (size_t)(wrow + 2 * hi) * NCH + ch;
  float*       q = yb + (size_t)(wrow + 8 * hi) * NCH + ch;

#pragma unroll
  for (int it = 0; it < NTILES / 8; ++it) {
    const int r = 4 * it * ROWSTEP;                   // compile-time row offset

    v8f acc = {};
#pragma unroll
    for (int s = 0; s < NSTEPS; ++s) {
      // B layout per k-step: VGPR0 -> K=4s+2*hi, VGPR1 -> K=4s+2*hi+1
      v2f B = { p[(r + 4 * s + 0) * NCH], p[(r + 4 * s + 1) * NCH] };
      acc = __builtin_amdgcn_wmma_f32_16x16x4_f32(
          false, A[s], false, B, (short)0, acc, false, false);
    }

    // D rows: VGPR v -> tile row v + 8*hi. Validity splits by lane half ->
    // one divergent straight-line region per tile. Output is never re-read:
    // non-temporal stores keep L2 for the cross-chunk padded B reads.
    if (MODE == 0) {            // valid tile rows 0..11
#pragma unroll
      for (int v = 0; v < 4; ++v) __builtin_nontemporal_store(acc[v], q + (r + v) * NCH);
      if (!hi) {
#pragma unroll
        for (int v = 4; v < 8; ++v) __builtin_nontemporal_store(acc[v], q + (r + v) * NCH);
      }
    } else if (MODE == 1) {     // valid tile rows 0..9
#pragma unroll
      for (int v = 0; v < 2; ++v) __builtin_nontemporal_store(acc[v], q + (r + v) * NCH);
      if (!hi) {
#pragma unroll
        for (int v = 2; v < 8; ++v) __builtin_nontemporal_store(acc[v], q + (r + v) * NCH);
      }
    } else {                    // valid tile rows 2..15
#pragma unroll
      for (int v = 2; v < 8; ++v) __builtin_nontemporal_store(acc[v], q + (r + v) * NCH);
      if (hi) {
#pragma unroll
        for (int v = 0; v < 2; ++v) __builtin_nontemporal_store(acc[v], q + (r + v) * NCH);
      }
    }
  }
}

__global__ __launch_bounds__(256) void wigner_apply(
    const float* __restrict__ x, const float* __restrict__ Dws,
    float* __restrict__ y)
{
  __shared__ float sD[3][16][16];
  const int n   = blockIdx.x;
  const int tid = threadIdx.x;
  const float* Dn = Dws + (size_t)n * 84;

  // Expand compact D's into padded block-diagonal 16x16 tiles (768 entries).
  for (int i = tid; i < 768; i += 256) {
    int tile = i >> 8, r = (i >> 4) & 15, c = i & 15;
    float v = 0.f;
    if (tile == 0) {
      if (r < 12 && c < 12 && (r / 3) == (c / 3)) v = Dn[0 + (r % 3) * 3 + (c % 3)];
    } else if (tile == 1) {
      if (r < 10 && c < 10 && (r / 5) == (c / 5)) v = Dn[9 + (r % 5) * 5 + (c % 5)];
    } else {                    // shifted +2: rows/cols 2..15 hold D3 x2
      int rr = r - 2, cc = c - 2;
      if (rr >= 0 && cc >= 0 && (rr / 7) == (cc / 7)) v = Dn[34 + (rr % 7) * 7 + (cc % 7)];
    }
    sD[tile][r][c] = v;
  }
  __syncthreads();

  const size_t base = (size_t)n * (DIM * NCH);
  const float* xb = x + base;
  float*       yb = y + base;

  // l = 0 section: D = 1 -> vectorized copy of rows 0..255 (2048 x b128).
  {
    const v4f* x4 = (const v4f*)xb;
    v4f*       y4 = (v4f*)yb;
#pragma unroll
    for (int i = 0; i < 8; ++i) {
      v4f v = x4[tid + i * 256];
      __builtin_nontemporal_store(v, y4 + tid + i * 256);
    }
  }

  const int wave  = tid >> 5;
  const int lane  = tid & 31;
  const int laneN = lane & 15;
  const int hi    = lane >> 4;

  run_section<0, 0, 128, 12, 3>(sD, xb, yb,  256,     wave, laneN, hi);  // l=1
  run_section<1, 1, 128, 10, 3>(sD, xb, yb, 1024,     wave, laneN, hi);  // l=2
  run_section<2, 2,  64, 14, 4>(sD, xb, yb, 1664 - 2, wave, laneN, hi);  // l=3
}

// ---------------------------------------------------------------------------
extern "C" void kernel_launch(void* const* d_in, const int* in_sizes, int n_in,
                              void* d_out, int out_size, void* d_ws, size_t ws_size,
                              hipStream_t stream) {
  const float* x     = (const float*)d_in[0];
  const float* alpha = (const float*)d_in[1];
  const float* beta  = (const float*)d_in[2];
  const float* gamma = (const float*)d_in[3];
  float* y   = (float*)d_out;
  float* Dws = (float*)d_ws;   // needs 1024*84*4 = 344064 bytes

  wigner_build_D<<<NB / 256, 256, 0, stream>>>(alpha, beta, gamma, Dws);
  wigner_apply<<<NB, 256, 0, stream>>>(x, Dws, y);
}